// GNN_24026047053899
// MI455X (gfx1250) — compile-verified
//
#include <hip/hip_runtime.h>

// GraphSAGE 2-layer forward for MI455X (gfx1250, wave32).
//
// Pipeline (memory-bound on edge gather/scatter; GEMMs are L2/LDS-fed WMMA):
//  1. deg[]  : float atomic in-degree histogram        (global_atomic_add_f32)
//  2. agg1[] : scatter-sum of x[src] rows into dst     (1 wave/edge, float4 gather)
//  3. layer1 : h = relu([agg1*inv_deg | x] @ [W1l;W1r] + b1)   (fp32 WMMA, K=256)
//  4. agg2[] : scatter-sum of h[src]
//  5. layer2 : out = [agg2*inv_deg | h] @ [W2l;W2r] + b2       (NCOLS=64)
//
// LDS weight layout is K-pair interleaved: element (k,n) lives at
//   lds[(k>>1)*PROW + 2n + (k&1)],  PROW = 2*NCOLS + 32
// so a lane's WMMA B-fragment {W[kb][n], W[kb+1][n]} is ONE ds_load_b64, and
// PROW % 64 == 32 puts the two lane-halves (pairs p / p+1) in disjoint bank
// windows. A-fragments are software-pipelined float2 global loads.
//
// Workspace (floats): deg[N] | agg1[N*128] | agg2[N*128] | h[N*128]  (~77 MB)
// edge_index treated as int32 per harness convention.

typedef float v2f __attribute__((ext_vector_type(2)));
typedef float v8f __attribute__((ext_vector_type(8)));

#define KDIM 128  // IN == HID == 128

__global__ void deg_kernel(const int* __restrict__ ei, float* __restrict__ deg, int E) {
  int e = blockIdx.x * blockDim.x + threadIdx.x;
  if (e < E) atomicAdd(&deg[ei[E + e]], 1.0f);
}

// One wave per edge; each lane moves 4 consecutive floats.
__global__ void scatter_kernel(const int* __restrict__ ei, const float* __restrict__ feat,
                               float* __restrict__ agg, int E) {
  long gtid = (long)blockIdx.x * blockDim.x + threadIdx.x;
  int e = (int)(gtid >> 5);
  if (e >= E) return;
  int lane = (int)(gtid & 31);
  int s = ei[e];
  int d = ei[E + e];
  const float4 v = *(const float4*)(feat + (size_t)s * KDIM + lane * 4);
  float* a = agg + (size_t)d * KDIM + lane * 4;
  atomicAdd(a + 0, v.x);
  atomicAdd(a + 1, v.y);
  atomicAdd(a + 2, v.z);
  atomicAdd(a + 3, v.w);
}

// Fused SAGE layer as one K=256 GEMM: [agg*inv_deg | x] (16x256) @ [Wl;Wr] (256xNCOLS).
// Block = 256 threads = 8 waves; each wave owns a 16-node row tile (128 nodes/block).
//
// WMMA fp32 16x16x4 fragment mapping (ISA 7.12.2, 32-bit layouts):
//   A:   lane L: m=L&15, VGPRs hold K = k0+2*(L>>4), +1  -> one float2 global load
//   B:   lane L: n=L&15, VGPRs hold rows kb, kb+1 of W   -> one ds_load_b64 (interleaved)
//   C/D: VGPR r: lanes 0-15 -> M=r, lanes 16-31 -> M=r+8, N = (L&15) + 16*nt
template <int NCOLS, bool RELU>
__launch_bounds__(256) __global__
void sage_layer(const float* __restrict__ agg, const float* __restrict__ xin,
                const float* __restrict__ deg, const float* __restrict__ Wl,
                const float* __restrict__ Wr, const float* __restrict__ bias,
                float* __restrict__ out, int nnodes) {
  constexpr int NT = NCOLS / 16;         // 16-wide output column tiles
  constexpr int PROW = 2 * NCOLS + 32;   // dwords per K-pair row (PROW%64==32)
  extern __shared__ __align__(16) float lds[];  // [128 pairs][PROW]
  const int tid = threadIdx.x;

  // ---- stage [Wl ; Wr] into pair-interleaved LDS (float4 global, scalar LDS) ----
  constexpr int QPR = NCOLS / 4;  // float4s per weight row
  for (int idx = tid; idx < 256 * QPR; idx += 256) {
    int row = idx / QPR;
    int q = idx - row * QPR;
    const float* srcw = (row < 128) ? (Wl + (size_t)row * NCOLS)
                                    : (Wr + (size_t)(row - 128) * NCOLS);
    float4 v = *(const float4*)(srcw + q * 4);
    float* p = &lds[(size_t)(row >> 1) * PROW + (q * 4) * 2 + (row & 1)];
    p[0] = v.x; p[2] = v.y; p[4] = v.z; p[6] = v.w;
  }
  __syncthreads();

  const int wave = tid >> 5;
  const int lane = tid & 31;
  const int m = lane & 15;
  const int hi = lane >> 4;
  const int node_base = blockIdx.x * 128 + wave * 16;
  if (node_base >= nnodes) return;  // whole-wave exit keeps EXEC all-ones for WMMA

  // Per-lane A row (clamped on the ragged tail; bogus rows dropped at store).
  int arow = node_base + m;
  if (arow >= nnodes) arow = nnodes - 1;
  const float scale = 1.0f / fmaxf(deg[arow], 1.0f);  // mean = sum * inv_deg
  const float* __restrict__ aggrow = agg + (size_t)arow * KDIM;
  const float* __restrict__ xrow = xin + (size_t)arow * KDIM;

  // Accumulators with bias folded into initial C.
  v8f acc[NT];
#pragma unroll
  for (int nt = 0; nt < NT; ++nt) {
    float b = bias[nt * 16 + m];
#pragma unroll
    for (int i = 0; i < 8; ++i) acc[nt][i] = b;
  }

  const int koff = 2 * hi;                       // lane-half K offset
  float2 nv = *(const float2*)(aggrow + koff);   // pipelined A prefetch

  // ---- K = 0..127 : aggregated-neighbor half (scaled by inv_deg) ----
#pragma unroll 4
  for (int k0 = 0; k0 < KDIM; k0 += 4) {
    float2 va = nv;
    const float* np = (k0 + 4 < KDIM) ? (aggrow + k0 + 4 + koff) : (xrow + koff);
    nv = *(const float2*)np;  // prefetch next step (or first step of half 2)
    v2f afrag;
    afrag[0] = va.x * scale;
    afrag[1] = va.y * scale;
    const v2f* wp = (const v2f*)&lds[(size_t)((k0 >> 1) + hi) * PROW];
    v2f bf[NT];
#pragma unroll
    for (int nt = 0; nt < NT; ++nt) bf[nt] = wp[m + nt * 16];  // ds_load_b64 each
#pragma unroll
    for (int nt = 0; nt < NT; ++nt)
      acc[nt] = __builtin_amdgcn_wmma_f32_16x16x4_f32(
          false, afrag, false, bf[nt], (short)0, acc[nt], false, false);
  }

  // ---- K = 128..255 : root-feature half ----
#pragma unroll 4
  for (int k0 = 0; k0 < KDIM; k0 += 4) {
    float2 va = nv;
    const float* np = (k0 + 4 < KDIM) ? (xrow + k0 + 4 + koff) : (xrow + koff);
    nv = *(const float2*)np;  // last prefetch is a harmless dummy
    v2f afrag;
    afrag[0] = va.x;
    afrag[1] = va.y;
    const v2f* wp = (const v2f*)&lds[(size_t)(((KDIM + k0) >> 1) + hi) * PROW];
    v2f bf[NT];
#pragma unroll
    for (int nt = 0; nt < NT; ++nt) bf[nt] = wp[m + nt * 16];
#pragma unroll
    for (int nt = 0; nt < NT; ++nt)
      acc[nt] = __builtin_amdgcn_wmma_f32_16x16x4_f32(
          false, afrag, false, bf[nt], (short)0, acc[nt], false, false);
  }

  // ---- store D tiles (optionally ReLU) ----
#pragma unroll
  for (int nt = 0; nt < NT; ++nt) {
    const int col = nt * 16 + m;
#pragma unroll
    for (int r = 0; r < 8; ++r) {
      int node = node_base + r + 8 * hi;
      if (node < nnodes) {
        float v = acc[nt][r];
        if (RELU) v = fmaxf(v, 0.0f);
        out[(size_t)node * NCOLS + col] = v;
      }
    }
  }
}

extern "C" void kernel_launch(void* const* d_in, const int* in_sizes, int n_in,
                              void* d_out, int out_size, void* d_ws, size_t ws_size,
                              hipStream_t stream) {
  const float* x   = (const float*)d_in[0];   // [N,128]
  const int*   ei  = (const int*)d_in[1];     // [2,E] (src row, then dst row)
  const float* W1l = (const float*)d_in[2];   // [128,128]
  const float* W1r = (const float*)d_in[3];   // [128,128]
  const float* b1  = (const float*)d_in[4];   // [128]
  const float* W2l = (const float*)d_in[5];   // [128,64]
  const float* W2r = (const float*)d_in[6];   // [128,64]
  const float* b2  = (const float*)d_in[7];   // [64]
  float* out = (float*)d_out;                 // [N,64]

  const int N = in_sizes[0] / KDIM;
  const int E = in_sizes[1] / 2;

  // Workspace carve-up: deg | agg1 | agg2 | h  (deg..agg2 contiguous -> one memset)
  float* ws   = (float*)d_ws;
  float* deg  = ws;
  float* agg1 = deg + N;
  float* agg2 = agg1 + (size_t)N * KDIM;
  float* h    = agg2 + (size_t)N * KDIM;
  (void)ws_size;  // requires N*(1+3*128)*4 bytes (~77 MB for N=50000)

  // Zero deg + agg1 + agg2 (h is fully overwritten by layer 1).
  hipMemsetAsync(deg, 0, (size_t)N * (1 + 2 * KDIM) * sizeof(float), stream);

  // Degrees.
  deg_kernel<<<(E + 255) / 256, 256, 0, stream>>>(ei, deg, E);

  // Layer-1 aggregation: agg1 = scatter_sum(x[src] -> dst)
  int sblocks = (int)(((long)E * 32 + 255) / 256);
  scatter_kernel<<<sblocks, 256, 0, stream>>>(ei, x, agg1, E);

  // Layer 1 fused GEMM + ReLU.
  int lblocks = (N + 127) / 128;
  size_t lds1 = (size_t)128 * (2 * 128 + 32) * sizeof(float);  // 144 KB
  hipFuncSetAttribute((const void*)&sage_layer<128, true>,
                      hipFuncAttributeMaxDynamicSharedMemorySize, (int)lds1);
  sage_layer<128, true><<<lblocks, 256, lds1, stream>>>(agg1, x, deg, W1l, W1r, b1, h, N);

  // Layer-2 aggregation: agg2 = scatter_sum(h[src] -> dst)
  scatter_kernel<<<sblocks, 256, 0, stream>>>(ei, h, agg2, E);

  // Layer 2 fused GEMM (no activation), writes d_out.
  size_t lds2 = (size_t)128 * (2 * 64 + 32) * sizeof(float);  // 80 KB
  hipFuncSetAttribute((const void*)&sage_layer<64, false>,
                      hipFuncAttributeMaxDynamicSharedMemorySize, (int)lds2);
  sage_layer<64, false><<<lblocks, 256, lds2, stream>>>(agg2, h, deg, W2l, W2r, b2, out, N);
}